// SelfAttention_49624052138278
// MI455X (gfx1250) — compile-verified
//
#include <hip/hip_runtime.h>

// ---------------------------------------------------------------------------
// SelfAttention (B=8, S=2048, D=512) for MI455X / gfx1250.
//
// The reference's attention einsum 'bqk,bvd->bqd' has no shared index:
//   attention[b,q,d] = (sum_k softmax_row)[b,q] * (sum_s V[b,s,d])
// and sum_k softmax == 1, so the output is
//   out[b,s,e] = (xsum[b,:] @ Wv^T + S*bv) @ Wfc^T + bfc   (broadcast over s)
// with xsum[b,d] = sum_s x[b,s,d].  This is HBM-roofline bound:
// read x (33.5MB) + write out (33.5MB) ~= 2.9us at 23.3 TB/s.
// The two 16x512x512 GEMMs run on v_wmma_f32_16x16x4_f32 (exact f32).
//
// All intermediate M=8 matrices are zero-padded to M=16 in workspace so the
// WMMA kernel has no divergent lanes anywhere (EXEC stays all-ones, loads are
// unconditional b64).
// ---------------------------------------------------------------------------

typedef float v2f __attribute__((ext_vector_type(2)));
typedef float v8f __attribute__((ext_vector_type(8)));

#define B_ 8
#define S_ 2048
#define D_ 512
#define CHUNKS 32
#define ROWS_PER_CHUNK (S_ / CHUNKS)   // 64

// --------------------------------------------------------------------------
// Kernel 1: partial column sums of x over 64-row chunks.
// grid = B*CHUNKS = 256 blocks, block = 128 threads, one float4 per thread.
// Each s-iteration reads one contiguous 2KB row per block via b128 loads.
// Data is single-use => direct global loads saturate HBM; LDS/TDM staging
// would only add latency.
// --------------------------------------------------------------------------
__global__ void k_partial_colsum(const float* __restrict__ x,
                                 float* __restrict__ partial) {
    const int bc    = blockIdx.x;            // b*CHUNKS + chunk
    const int b     = bc / CHUNKS;
    const int chunk = bc % CHUNKS;
    const int d4    = threadIdx.x;           // float4 index within row (0..127)
    const float4* p = (const float4*)(x + ((size_t)b * S_ +
                         (size_t)chunk * ROWS_PER_CHUNK) * D_) + d4;
    float4 acc = make_float4(0.f, 0.f, 0.f, 0.f);
#pragma unroll 8
    for (int s = 0; s < ROWS_PER_CHUNK; ++s) {
        const float4 v = p[(size_t)s * (D_ / 4)];
        acc.x += v.x; acc.y += v.y; acc.z += v.z; acc.w += v.w;
    }
    ((float4*)(partial + (size_t)bc * D_))[d4] = acc;
}

// --------------------------------------------------------------------------
// Kernel 2: reduce the 32 chunk partials -> xsum16[16][512] (rows 8..15 = 0).
// grid = 16 blocks x 512 threads.  Fixed summation order => deterministic.
// --------------------------------------------------------------------------
__global__ void k_final_colsum(const float* __restrict__ partial,
                               float* __restrict__ xsum16) {
    const int m = blockIdx.x;                // 0..15 (padded row)
    const int d = threadIdx.x;
    float acc = 0.0f;
    if (m < B_) {
#pragma unroll
        for (int c = 0; c < CHUNKS; ++c)
            acc += partial[((size_t)m * CHUNKS + c) * D_ + d];
    }
    xsum16[(size_t)m * D_ + d] = acc;
}

// --------------------------------------------------------------------------
// Kernel 3: out16[m,e] = sum_k A16[m,k] * W[e,k] + beta*bias[e] for m<8,
//           out16[m,e] = 0 for m in 8..15 (keeps padding for the next GEMM).
// One wave per 16-wide e-tile (grid = 32 blocks x 32 threads).
// K chained through the f32 WMMA accumulator in 128 steps of K=4.
//
// Fragment layouts per CDNA5 ISA 7.12.2 (wave32):
//   A (16x4 f32):  lane<16 -> row M=lane,   {K=k0, k0+1}
//                  lane>=16 -> row M=lane-16, {K=k0+2, k0+3}
//   B (4x16 f32):  lane<16 -> col N=lane,   {B[k0][N], B[k0+1][N]}
//                  lane>=16 -> col N=lane-16, {B[k0+2][N], B[k0+3][N]}
//   C/D (16x16):   VGPR r: lanes 0-15 -> M=r, lanes 16-31 -> M=r+8
// No divergent control flow anywhere: EXEC is all-ones at every WMMA.
// --------------------------------------------------------------------------
__global__ void k_gemm_wmma(const float* __restrict__ A16,  // [16][512], rows 8+ zero
                            const float* __restrict__ W,    // [512][512] row-major
                            const float* __restrict__ bias, // [512]
                            float beta,
                            float* __restrict__ out16) {    // [16][512]
    const int lane  = threadIdx.x;       // 0..31
    const int half  = lane >> 4;         // 0 or 1
    const int l16   = lane & 15;
    const int etile = blockIdx.x;        // 0..31
    const int eCol  = etile * 16 + l16;  // this lane's B-fragment column

    const float* aRow = A16 + (size_t)l16 * D_;   // pad rows are real zeros
    const float* wRow = W + (size_t)eCol * D_;

    v8f c = {};                          // zero accumulator
#pragma unroll 8
    for (int k0 = 0; k0 < D_; k0 += 4) {
        const int ka = k0 + half * 2;    // 8-byte aligned offsets
        const v2f a  = *(const v2f*)(aRow + ka);
        const v2f bf = *(const v2f*)(wRow + ka);
        // (neg_a, A, neg_b, B, c_mod, C, reuse_a, reuse_b)
        c = __builtin_amdgcn_wmma_f32_16x16x4_f32(
                false, a, false, bf, (short)0, c, false, false);
    }

    // All 32 lanes store: lane half 0 -> rows 0..7 (valid, add bias),
    // lane half 1 -> rows 8..15 (write zero padding).  Pure v_cndmask select.
    const float bb   = beta * bias[eCol];
    const float keep = (half == 0) ? 1.0f : 0.0f;
#pragma unroll
    for (int r = 0; r < 8; ++r) {
        const int m = r + half * 8;
        out16[(size_t)m * D_ + eCol] = keep * (c[r] + bb);
    }
}

// --------------------------------------------------------------------------
// Kernel 4: out[b,s,:] = row[b,:] broadcast.  float4 stores, write-bound.
// row[] (16KB live portion) stays hot in L2 across the whole grid.
// --------------------------------------------------------------------------
__global__ void k_broadcast(const float* __restrict__ rowv,   // [16][512], rows 0..7 live
                            float* __restrict__ out) {
    const size_t i  = (size_t)blockIdx.x * blockDim.x + threadIdx.x; // float4 idx
    const int    e4 = (int)(i & (D_ / 4 - 1));                       // 0..127
    const size_t r  = i >> 7;                                        // b*S + s
    const int    b  = (int)(r >> 11);                                // /S_
    const float4 v  = ((const float4*)rowv)[(size_t)b * (D_ / 4) + e4];
    ((float4*)out)[i] = v;
}

// --------------------------------------------------------------------------
// Launch.  Inputs: 0:x 1:Wq 2:bq 3:Wk 4:bk 5:Wv 6:bv 7:Wfc 8:bfc
// ws layout (floats):
//   partial[8*32*512] | xsum16[16*512] | vsum16[16*512] | row16[16*512]
// Every ws word is written before it is read => no zero-init, no cross-call
// state, no atomics => fully deterministic.
// --------------------------------------------------------------------------
extern "C" void kernel_launch(void* const* d_in, const int* in_sizes, int n_in,
                              void* d_out, int out_size, void* d_ws, size_t ws_size,
                              hipStream_t stream) {
    (void)in_sizes; (void)n_in; (void)out_size; (void)ws_size;

    const float* x   = (const float*)d_in[0];
    const float* Wv  = (const float*)d_in[5];
    const float* bv  = (const float*)d_in[6];
    const float* Wfc = (const float*)d_in[7];
    const float* bfc = (const float*)d_in[8];
    float* out = (float*)d_out;

    float* ws      = (float*)d_ws;
    float* partial = ws;                                   // 512 KB
    float* xsum16  = partial + (size_t)B_ * CHUNKS * D_;   // 32 KB
    float* vsum16  = xsum16 + (size_t)16 * D_;             // 32 KB
    float* row16   = vsum16 + (size_t)16 * D_;             // 32 KB

    k_partial_colsum<<<B_ * CHUNKS, D_ / 4, 0, stream>>>(x, partial);
    k_final_colsum<<<16, D_, 0, stream>>>(partial, xsum16);
    // Vsum = xsum @ Wv^T + S*bv   (bias folded: sum over S of bv)
    k_gemm_wmma<<<D_ / 16, 32, 0, stream>>>(xsum16, Wv, bv, (float)S_, vsum16);
    // row  = Vsum @ Wfc^T + bfc   (softmax row-sum == 1 absorbed here)
    k_gemm_wmma<<<D_ / 16, 32, 0, stream>>>(vsum16, Wfc, bfc, 1.0f, row16);

    const size_t nf4 = (size_t)B_ * S_ * D_ / 4;           // 2,097,152
    k_broadcast<<<(unsigned)(nf4 / 256), 256, 0, stream>>>(row16, out);
}